// Product3body_70480413328172
// MI455X (gfx1250) — compile-verified
//
#include <hip/hip_runtime.h>
#include <math.h>

typedef float v2f __attribute__((ext_vector_type(2)));
typedef float v8f __attribute__((ext_vector_type(8)));

#define C_CH 128
#define IRR  9

// ---------------------------------------------------------------------------
// Real Clebsch-Gordan table generation on device (exact mirror of reference).
// Layout (float offsets in ws):
//  (l1,l2,l3): (0,0,0)@0 (1,1,0)@1 (2,2,0)@10 (0,1,1)@35 (1,0,1)@44
//  (1,2,1)@53 (2,1,1)@98 (0,2,2)@143 (2,0,2)@168 (1,1,2)@193 (2,2,2)@238
// Each stored as cg[off + (b*d2 + d)*d3 + a], b:l1-idx, d:l2-idx, a:l3-idx.
// ---------------------------------------------------------------------------
struct cplx { double re, im; };
__device__ inline cplx cmul(cplx a, cplx b){ return {a.re*b.re - a.im*b.im, a.re*b.im + a.im*b.re}; }

__device__ double dfact(int n){ double r = 1.0; for (int i = 2; i <= n; ++i) r *= (double)i; return r; }

__device__ double cg_complex(int l1,int l2,int l3,int m1,int m2){
  int m3 = m1 + m2;
  if (m3 < -l3 || m3 > l3) return 0.0;
  double pref = sqrt((double)(2*l3+1) * dfact(l3+l1-l2) * dfact(l3-l1+l2)
                     * dfact(l1+l2-l3) / dfact(l1+l2+l3+1));
  pref *= sqrt(dfact(l3+m3)*dfact(l3-m3)*dfact(l1-m1)*dfact(l1+m1)
               *dfact(l2-m2)*dfact(l2+m2));
  int kmin = 0;
  if (l2-l3-m1 > kmin) kmin = l2-l3-m1;
  if (l1+m2-l3 > kmin) kmin = l1+m2-l3;
  int kmax = l1+l2-l3;
  if (l1-m1 < kmax) kmax = l1-m1;
  if (l2+m2 < kmax) kmax = l2+m2;
  double s = 0.0;
  for (int k = kmin; k <= kmax; ++k){
    double den = dfact(k)*dfact(l1+l2-l3-k)*dfact(l1-m1-k)*dfact(l2+m2-k)
                 *dfact(l3-l2+m1+k)*dfact(l3-l1-m2+k);
    s += ((k & 1) ? -1.0 : 1.0) / den;
  }
  return pref * s;
}

// Real<->complex SH change of basis, U[row,col], row/col in [0, 2l]
__device__ cplx Uent(int l, int r, int c){
  const double is2 = 0.7071067811865475244;
  int mr = r - l;
  if (mr == 0) return { (c == l) ? 1.0 : 0.0, 0.0 };
  if (mr > 0){
    int m = mr; double sgn = (m & 1) ? -1.0 : 1.0;
    if (c == l+m) return { sgn*is2, 0.0 };
    if (c == l-m) return { is2, 0.0 };
    return {0.0, 0.0};
  }
  int m = -mr; double sgn = (m & 1) ? -1.0 : 1.0;
  if (c == l+m) return { 0.0, -sgn*is2 };
  if (c == l-m) return { 0.0, is2 };
  return {0.0, 0.0};
}

__global__ void cg_init_kernel(float* __restrict__ cg){
  if (threadIdx.x != 0 || blockIdx.x != 0) return;
  for (int i = 0; i < 512; ++i) cg[i] = 0.0f;
  const int L1[11]  = {0,1,2, 0,1,1,2, 0,2,1,2};
  const int L2[11]  = {0,1,2, 1,0,2,1, 2,0,1,2};
  const int L3[11]  = {0,0,0, 1,1,1,1, 2,2,2,2};
  const int OFF[11] = {0,1,10, 35,44,53,98, 143,168,193,238};
  for (int p = 0; p < 11; ++p){
    int l1 = L1[p], l2 = L2[p], l3 = L3[p], off = OFF[p];
    int d1 = 2*l1+1, d2 = 2*l2+1, d3 = 2*l3+1;
    for (int b = 0; b < d1; ++b)
      for (int d = 0; d < d2; ++d)
        for (int a = 0; a < d3; ++a){
          double s = 0.0;
          for (int i = 0; i < d1; ++i)
            for (int j = 0; j < d2; ++j){
              int k = (i - l1) + (j - l2) + l3;         // m3 = m1 + m2
              if (k < 0 || k >= d3) continue;
              double cc = cg_complex(l1, l2, l3, i - l1, j - l2);
              if (cc == 0.0) continue;
              cplx u1 = Uent(l1, b, i); u1.im = -u1.im; // conj
              cplx u2 = Uent(l2, d, j); u2.im = -u2.im; // conj
              cplx u3 = Uent(l3, a, k);
              cplx t = cmul(cmul(u1, u2), u3);
              s += t.re * cc;
            }
          cg[off + (b*d2 + d)*d3 + a] = (float)s;
        }
  }
}

// ---------------------------------------------------------------------------
// Weight transpose: w[mat][c][d] -> wT[mat][d][c], so a B-fragment lane read
// (two consecutive K values at a fixed output column) is one contiguous float2.
// ---------------------------------------------------------------------------
__global__ void transpose_w_kernel(const float* __restrict__ wlin,
                                   const float* __restrict__ wfctp,
                                   float* __restrict__ wlinT,
                                   float* __restrict__ wfctpT){
  int idx = blockIdx.x * blockDim.x + threadIdx.x;
  if (idx < 3*C_CH*C_CH){
    int mat = idx >> 14, rem = idx & 16383, cc = rem >> 7, dd = rem & 127;
    wlinT[(mat << 14) + (dd << 7) + cc] = wlin[idx];
  }
  if (idx < 12*C_CH*C_CH){
    int mat = idx >> 14, rem = idx & 16383, cc = rem >> 7, dd = rem & 127;
    wfctpT[(mat << 14) + (dd << 7) + cc] = wfctp[idx];
  }
}

// ---------------------------------------------------------------------------
// Per-(n,c) equivariant products x2 = x1(x)x1, x3 = x2(x)x1, body mixing.
// ---------------------------------------------------------------------------
__device__ void contract3j(const float* __restrict__ cg,
                           const float* __restrict__ X,
                           const float* __restrict__ Y,
                           float* __restrict__ O){
  // l3 = 0: paths (0,0),(1,1),(2,2); norm 1/sqrt(3)
  {
    float s = X[0]*Y[0]*cg[0];
    #pragma unroll
    for (int i = 0; i < 3; ++i)
      #pragma unroll
      for (int j = 0; j < 3; ++j)
        s += X[1+i]*Y[1+j]*cg[1 + i*3 + j];
    #pragma unroll
    for (int i = 0; i < 5; ++i)
      #pragma unroll
      for (int j = 0; j < 5; ++j)
        s += X[4+i]*Y[4+j]*cg[10 + i*5 + j];
    O[0] = s * 0.57735026918962576f;
  }
  // l3 = 1: paths (0,1),(1,0),(1,2),(2,1); norm 1/2
  #pragma unroll
  for (int a = 0; a < 3; ++a){
    float s = 0.0f;
    #pragma unroll
    for (int j = 0; j < 3; ++j) s += X[0]*Y[1+j]*cg[35 + j*3 + a];
    #pragma unroll
    for (int i = 0; i < 3; ++i) s += X[1+i]*Y[0]*cg[44 + i*3 + a];
    #pragma unroll
    for (int i = 0; i < 3; ++i)
      #pragma unroll
      for (int j = 0; j < 5; ++j)
        s += X[1+i]*Y[4+j]*cg[53 + (i*5+j)*3 + a];
    #pragma unroll
    for (int i = 0; i < 5; ++i)
      #pragma unroll
      for (int j = 0; j < 3; ++j)
        s += X[4+i]*Y[1+j]*cg[98 + (i*3+j)*3 + a];
    O[1+a] = s * 0.5f;
  }
  // l3 = 2: paths (0,2),(2,0),(1,1),(2,2); norm 1/2
  #pragma unroll
  for (int a = 0; a < 5; ++a){
    float s = 0.0f;
    #pragma unroll
    for (int j = 0; j < 5; ++j) s += X[0]*Y[4+j]*cg[143 + j*5 + a];
    #pragma unroll
    for (int i = 0; i < 5; ++i) s += X[4+i]*Y[0]*cg[168 + i*5 + a];
    #pragma unroll
    for (int i = 0; i < 3; ++i)
      #pragma unroll
      for (int j = 0; j < 3; ++j)
        s += X[1+i]*Y[1+j]*cg[193 + (i*3+j)*5 + a];
    #pragma unroll
    for (int i = 0; i < 5; ++i)
      #pragma unroll
      for (int j = 0; j < 5; ++j)
        s += X[4+i]*Y[4+j]*cg[238 + (i*5+j)*5 + a];
    O[4+a] = s * 0.5f;
  }
}

// b-matrix rows grouped by l:  row(l=0) = n           in [0, N)
//                              row(l=1) = N  + 3n + m in [N, 4N)
//                              row(l=2) = 4N + 5n + m in [4N, 9N)
__global__ __launch_bounds__(C_CH) void body_kernel(
    const float* __restrict__ nf, const float* __restrict__ wbody,
    const float* __restrict__ cg_g, float* __restrict__ bmat, int N){
  __shared__ float cg[368];
  for (int i = threadIdx.x; i < 363; i += C_CH) cg[i] = cg_g[i];
  __syncthreads();
  const int n = blockIdx.x;
  const int c = threadIdx.x;
  if (n >= N) return;

  float x1[9], x2[9], x3[9];
  const float* p = nf + (size_t)n*(C_CH*IRR) + (size_t)c*IRR;
  #pragma unroll
  for (int i = 0; i < 9; ++i) x1[i] = p[i];

  contract3j(cg, x1, x1, x2);   // 2-body
  contract3j(cg, x2, x1, x3);   // 3-body

  const float rs3 = 0.57735026918962576f;  // 1/sqrt(3) (linear_body norm)
  {
    float bb = (wbody[0]*x1[0] + wbody[1]*x2[0] + wbody[2]*x3[0]) * rs3;
    bmat[(size_t)n * C_CH + c] = bb;
  }
  #pragma unroll
  for (int m = 0; m < 3; ++m){
    float bb = (wbody[3]*x1[1+m] + wbody[4]*x2[1+m] + wbody[5]*x3[1+m]) * rs3;
    bmat[((size_t)N + (size_t)n*3 + m) * C_CH + c] = bb;
  }
  #pragma unroll
  for (int m = 0; m < 5; ++m){
    float bb = (wbody[6]*x1[4+m] + wbody[7]*x2[4+m] + wbody[8]*x3[4+m]) * rs3;
    bmat[((size_t)4*N + (size_t)n*5 + m) * C_CH + c] = bb;
  }
}

// ---------------------------------------------------------------------------
// Fused GEMM chain: one wave per 16-row tile.
//  Stage 1: H(16x128) = Btile @ w_lin[l]          (256 x v_wmma_f32_16x16x4_f32)
//  LDS transpose C/D-layout -> A-layout
//  Stage 2: Out = sum_a diag(attr[:,a]/256) H @ w_fctp[l][a]   (1024 WMMAs)
//  (1/256 = 1/(sqrt(C)*sqrt(A*C)) folds both norms)
// ---------------------------------------------------------------------------
__global__ __launch_bounds__(128) void gemm_kernel(
    const float* __restrict__ bmat,  const float* __restrict__ wlinT,
    const float* __restrict__ wfctpT,const float* __restrict__ attr,
    const float* __restrict__ sc,    float* __restrict__ out,
    int N, int t0, int t1, int t2){
  __shared__ float hlds[4][16][132];           // +4 pad vs 128 to spread banks
  const int wave = threadIdx.x >> 5;
  const int lane = threadIdx.x & 31;
  const int half = lane >> 4;                  // K-half selector (f32 WMMA layout)
  const int r    = lane & 15;                  // row (A) / column (B) index

  const int tTot = t0 + t1 + t2;
  int t = blockIdx.x * 4 + wave;
  if (t >= tTot) t = tTot - 1;                 // redundant waves redo last tile

  int l, blockStart, rows, row0;
  if (t < t0)          { l = 0; blockStart = 0;   rows = N;   row0 = t*16; }
  else if (t < t0+t1)  { l = 1; blockStart = N;   rows = 3*N; row0 = (t-t0)*16; }
  else                 { l = 2; blockStart = 4*N; rows = 5*N; row0 = (t-t0-t1)*16; }

  int myRow = row0 + r;
  if (myRow >= rows) myRow = rows - 1;         // clamp loads; stores masked later

  const float* Abase = bmat + ((size_t)blockStart + (size_t)myRow) * C_CH;
  const float* Blin  = wlinT + (size_t)l * C_CH * C_CH;

  v8f acc[8] = {};
  #pragma unroll 4
  for (int k = 0; k < 32; ++k){
    const int kk = k*4 + half*2;               // A: lane(half,r) holds K=kk,kk+1
    v2f a = *(const v2f*)(Abase + kk);
    #pragma unroll
    for (int d = 0; d < 8; ++d){               // B: lane(half,r) holds col d*16+r
      v2f b = *(const v2f*)(Blin + (size_t)(d*16 + r) * C_CH + kk);
      acc[d] = __builtin_amdgcn_wmma_f32_16x16x4_f32(
          false, a, false, b, (short)0, acc[d], false, false);
    }
  }

  // C/D layout (row = v + 8*half, col = d*16 + r) -> LDS row-major tile
  #pragma unroll
  for (int d = 0; d < 8; ++d)
    #pragma unroll
    for (int v = 0; v < 8; ++v)
      hlds[wave][v + 8*half][d*16 + r] = acc[d][v];
  __syncthreads();

  const int dim  = 2*l + 1;
  const int offl = (l == 0) ? 0 : ((l == 1) ? 1 : 4);
  const int nMine = myRow / dim;               // node for this lane's A row
  float as[4];
  #pragma unroll
  for (int a = 0; a < 4; ++a) as[a] = attr[nMine*4 + a] * (1.0f/256.0f);

  v8f acc2[8] = {};
  const float* Bf0 = wfctpT + (size_t)l * 4 * C_CH * C_CH;
  for (int a = 0; a < 4; ++a){
    const float* Bf = Bf0 + (size_t)a * C_CH * C_CH;
    const float sa = as[a];
    #pragma unroll 4
    for (int k = 0; k < 32; ++k){
      const int kk = k*4 + half*2;
      v2f af;
      af.x = hlds[wave][r][kk]     * sa;       // A-layout read + per-row scale
      af.y = hlds[wave][r][kk + 1] * sa;
      #pragma unroll
      for (int d = 0; d < 8; ++d){
        v2f b = *(const v2f*)(Bf + (size_t)(d*16 + r) * C_CH + kk);
        acc2[d] = __builtin_amdgcn_wmma_f32_16x16x4_f32(
            false, af, false, b, (short)0, acc2[d], false, false);
      }
    }
  }

  // Epilogue: scatter to [N][C][9] layout, add sc
  #pragma unroll
  for (int d = 0; d < 8; ++d){
    #pragma unroll
    for (int v = 0; v < 8; ++v){
      int rr = row0 + v + 8*half;
      if (rr < rows){
        int nn = rr / dim, mm = rr - nn*dim;
        size_t oi = (size_t)nn*(C_CH*IRR) + (size_t)(d*16 + r)*IRR + offl + mm;
        out[oi] = acc2[d][v] + sc[oi];
      }
    }
  }
}

// ---------------------------------------------------------------------------
extern "C" void kernel_launch(void* const* d_in, const int* in_sizes, int n_in,
                              void* d_out, int out_size, void* d_ws, size_t ws_size,
                              hipStream_t stream){
  const float* nf    = (const float*)d_in[0];  // [N, C, 9]
  const float* attr  = (const float*)d_in[1];  // [N, 4]
  const float* sc    = (const float*)d_in[2];  // [N, C*9]
  const float* wbody = (const float*)d_in[3];  // [3, 3]
  const float* wlin  = (const float*)d_in[4];  // [3, C, C]
  const float* wfctp = (const float*)d_in[5];  // [3, 4, C, C]
  float* out = (float*)d_out;
  const int N = in_sizes[0] / (C_CH*IRR);

  float* ws     = (float*)d_ws;
  float* cg     = ws;                          // 512 floats
  float* wlinT  = ws + 512;                    // 3  * 128*128
  float* wfctpT = wlinT + 3*C_CH*C_CH;         // 12 * 128*128  (ends @246272)
  float* bmat   = ws + 262144;                 // 9*N*128 floats

  cg_init_kernel<<<1, 32, 0, stream>>>(cg);
  transpose_w_kernel<<<(12*C_CH*C_CH + 255)/256, 256, 0, stream>>>(
      wlin, wfctp, wlinT, wfctpT);
  body_kernel<<<N, C_CH, 0, stream>>>(nf, wbody, cg, bmat, N);

  const int t0 = (N + 15)/16, t1 = (3*N + 15)/16, t2 = (5*N + 15)/16;
  const int blocks = (t0 + t1 + t2 + 3)/4;
  gemm_kernel<<<blocks, 128, 0, stream>>>(bmat, wlinT, wfctpT, attr, sc, out,
                                          N, t0, t1, t2);
}